// One2ManyCtrlPointHungarianMatcher_withdynamic_28295244546873
// MI455X (gfx1250) — compile-verified
//
#include <hip/hip_runtime.h>
#include <hip/hip_bf16.h>

typedef __attribute__((ext_vector_type(2))) float v2f;
typedef __attribute__((ext_vector_type(8))) float v8f;

namespace {
constexpr int B = 4, Nq = 300, Nt = 32, P = 25, T = 25, V1 = 97;
constexpr int BLANK = V1 - 1;
constexpr int S = 2 * T + 1;   // 51
constexpr int MATCH = 5;
constexpr float NEGF = -1.0e9f;
constexpr float INFC = 1.0e8f;
constexpr float CLASS_W = 2.0f, COORD_W = 5.0f, TEXT_W = 4.0f;

constexpr int OUT_COST = 0;
constexpr int OUT_SRC  = B * Nq * Nt;              // 38400
constexpr int OUT_TGT  = OUT_SRC + B * Nt * MATCH; // +640
constexpr int OUT_MIN  = OUT_TGT + B * Nt * MATCH; // +640
constexpr int OUT_MASK = OUT_MIN + B * Nq;         // +1200
} // namespace

__device__ __forceinline__ float lae(float x, float y) {
  float m = fmaxf(x, y);
  return m + __logf(__expf(x - m) + __expf(y - m));
}

// ---------------- K1: focal class cost, mean over P ----------------
__global__ __launch_bounds__(256) void class_cost_kernel(
    const float* __restrict__ lg, float* __restrict__ cls) {
  int i = blockIdx.x * blockDim.x + threadIdx.x;
  if (i >= B * Nq) return;
  const float* p = lg + (size_t)i * P;
  float acc = 0.0f;
  #pragma unroll
  for (int j = 0; j < P; ++j) {
    float x = p[j];
    float s = 1.0f / (1.0f + __expf(-x));
    float pos = 0.25f * (1.0f - s) * (1.0f - s) * (-__logf(s + 1e-8f));
    float neg = 0.75f * s * s * (-__logf(1.0f - s + 1e-8f));
    acc += pos - neg;
  }
  cls[i] = acc / (float)P;
}

// ---------------- K2: log-softmax over V1 with WMMA sum-of-exp -----
// 16 rows per wave, single load phase: each lane preloads its A-tile
// partition (lane<16: K=4j,4j+1 ; lane>=16: K=4j+2,4j+3) once; those
// registers feed max, exp->WMMA chain, and the transposed store.
// Output transposed to (B,T,V1,Nq) for coalesced CTC gathers.
__global__ __launch_bounds__(256) void logsoftmax_wmma_kernel(
    const float* __restrict__ txt,   // (B,Nq,T,V1) row-major
    float* __restrict__ logT) {      // (B,T,V1,Nq)
  __shared__ float lsum[8][16];
  const int lane = threadIdx.x & 31;
  const int wv   = threadIdx.x >> 5;
  const int half = lane >> 4;       // K-half within A tile
  const int rl   = lane & 15;       // row within 16-row group
  const int R = B * Nq * T;

  int base = (blockIdx.x * 8 + wv) * 16;
  int row  = base + rl;
  bool valid = row < R;
  int rowc = valid ? row : (R - 1);
  int b = rowc / (Nq * T);
  int rem = rowc - b * (Nq * T);
  int q = rem / T;
  int t = rem - q * T;
  const float* rowp = txt + (size_t)rowc * V1;

  // ---- single load phase: 25 K-pairs per lane, tail clamped ----
  v2f a[25];
  #pragma unroll
  for (int j = 0; j < 25; ++j) {
    int k0 = 4 * j + 2 * half;
    int i0 = (k0     < V1) ? k0     : (V1 - 1);
    int i1 = (k0 + 1 < V1) ? k0 + 1 : (V1 - 1);
    float x0 = rowp[i0];
    float x1 = rowp[i1];
    a[j].x = (k0     < V1) ? x0 : -3.0e38f;   // pad -> exp() == 0
    a[j].y = (k0 + 1 < V1) ? x1 : -3.0e38f;
  }

  // ---- row max from registers + pair exchange ----
  float m = -3.0e38f;
  #pragma unroll
  for (int j = 0; j < 25; ++j) m = fmaxf(m, fmaxf(a[j].x, a[j].y));
  m = fmaxf(m, __shfl_xor(m, 16, 32));

  // ---- sum of exp via chained V_WMMA_F32_16X16X4_F32, B = ones ----
  v8f c = {0.f, 0.f, 0.f, 0.f, 0.f, 0.f, 0.f, 0.f};
  v2f ones; ones.x = 1.0f; ones.y = 1.0f;
  #pragma unroll
  for (int j = 0; j < 25; ++j) {
    v2f e;
    e.x = __expf(a[j].x - m);
    e.y = __expf(a[j].y - m);
    c = __builtin_amdgcn_wmma_f32_16x16x4_f32(false, e, false, ones,
                                              (short)0, c, false, false);
  }
  // D layout: lanes 0-15 hold rows 0..7 in c[0..7]; lanes 16-31 rows 8..15
  if (lane == 0) {
    #pragma unroll
    for (int r = 0; r < 8; ++r) lsum[wv][r] = c[r];
  }
  if (lane == 16) {
    #pragma unroll
    for (int r = 0; r < 8; ++r) lsum[wv][8 + r] = c[r];
  }
  __syncthreads();
  float lls = m + __logf(lsum[wv][rl]);   // logsumexp of my row

  // ---- transposed store straight from the preloaded registers ----
  if (valid) {
    float* outp = logT + ((size_t)(b * T + t) * V1) * Nq + q;
    #pragma unroll
    for (int j = 0; j < 25; ++j) {
      int k0 = 4 * j + 2 * half;
      if (k0     < V1) outp[(size_t)k0 * Nq]       = a[j].x - lls;
      if (k0 + 1 < V1) outp[(size_t)(k0 + 1) * Nq] = a[j].y - lls;
    }
  }
}

// ---------------- K3: CTC DP + keypoint L1 + cost assembly ----------
__global__ __launch_bounds__(320) void ctc_cost_kernel(
    const float* __restrict__ logT,     // (B,T,V1,Nq)
    const float* __restrict__ cls,      // (B*Nq)
    const float* __restrict__ predPts,  // (B,Nq,P,2)
    const float* __restrict__ tgtPts,   // (B,Nt,P,2)
    const int*   __restrict__ texts,    // (B,Nt,T)
    float* __restrict__ cost) {         // (B,Nq,Nt)
  __shared__ int   sExt[S];
  __shared__ int   sExtOff[S];
  __shared__ int   sSkip[S];
  __shared__ float sPts[2 * P];
  __shared__ int   sLen;

  int tid = threadIdx.x;
  int b  = blockIdx.x >> 5;
  int nt = blockIdx.x & 31;
  const int* tx = texts + (size_t)(b * Nt + nt) * T;
  if (tid < S) sExt[tid] = (tid & 1) ? tx[tid >> 1] : BLANK;
  if (tid < 2 * P) sPts[tid] = tgtPts[(size_t)(b * Nt + nt) * (2 * P) + tid];
  if (tid == 0) {
    int cN = 0;
    for (int j = 0; j < T; ++j) cN += (tx[j] != BLANK) ? 1 : 0;
    sLen = cN;
  }
  __syncthreads();
  if (tid < S) {
    sSkip[tid]   = (tid >= 2) && (sExt[tid] != BLANK) && (sExt[tid] != sExt[tid - 2]);
    sExtOff[tid] = sExt[tid] * Nq;
  }
  __syncthreads();
  int q = tid;
  if (q >= Nq) return;

  // L1 keypoint cost against LDS-staged target points
  const float* pp = predPts + (size_t)(b * Nq + q) * (2 * P);
  float kc = 0.0f;
  #pragma unroll
  for (int i = 0; i < 2 * P; ++i) kc += fabsf(pp[i] - sPts[i]);

  // skip conditions -> register bitmask
  unsigned long long skipm = 0ull;
  #pragma unroll
  for (int s = 0; s < S; ++s)
    if (sSkip[s]) skipm |= (1ull << s);

  // CTC alpha recursion, alpha kept in registers, in-place descending-s update
  float alpha[S];
  #pragma unroll
  for (int s = 0; s < S; ++s) alpha[s] = NEGF;
  const float* b0 = logT + ((size_t)(b * T) * V1) * Nq + q;
  alpha[0] = b0[sExtOff[0]];
  alpha[1] = b0[sExtOff[1]];

  for (int t = 1; t < T; ++t) {
    const float* bt = logT + ((size_t)(b * T + t) * V1) * Nq + q;
    if (t + 1 < T) __builtin_prefetch(bt + (size_t)V1 * Nq, 0, 1);
    #pragma unroll
    for (int s = S - 1; s >= 0; --s) {
      float x = alpha[s];
      float y = (s >= 1) ? alpha[s - 1] : NEGF;
      float z = (s >= 2) ? ((((skipm >> s) & 1ull) != 0ull) ? alpha[s - 2] : NEGF)
                         : NEGF;
      alpha[s] = lae(lae(x, y), z) + bt[sExtOff[s]];
    }
  }

  int l = sLen;
  float a1v = NEGF, a0v = NEGF;
  #pragma unroll
  for (int s = 0; s < S; ++s) {
    a1v = (s == 2 * l)     ? alpha[s] : a1v;
    a0v = (s == 2 * l - 1) ? alpha[s] : a0v;
  }
  float loss = -lae(a1v, a0v);
  if (loss > 1.0e6f) loss = 0.0f;
  loss = loss / (float)l;

  cost[(size_t)(b * Nq + q) * Nt + nt] =
      CLASS_W * cls[b * Nq + q] + COORD_W * kc + TEXT_W * loss;
}

// ---------------- K4: stable top-5 smallest cost per (b,nt) ---------
__global__ __launch_bounds__(128) void topk_kernel(
    const float* __restrict__ cost, float* __restrict__ outSrc,
    float* __restrict__ outTgt) {
  int gw   = (blockIdx.x * blockDim.x + threadIdx.x) >> 5;
  int lane = threadIdx.x & 31;
  int b  = gw >> 5;
  int nt = gw & 31;

  float v[10];
  #pragma unroll
  for (int i = 0; i < 10; ++i) {
    int q = lane + 32 * i;
    v[i] = (q < Nq) ? cost[(size_t)(b * Nq + q) * Nt + nt] : 3.0e38f;
  }
  for (int j = 0; j < MATCH; ++j) {
    float bv = 3.0e38f;
    int   bq = 0x7fffffff;
    #pragma unroll
    for (int i = 0; i < 10; ++i) {
      int q = lane + 32 * i;
      if (v[i] < bv) { bv = v[i]; bq = q; }
    }
    #pragma unroll
    for (int off = 16; off >= 1; off >>= 1) {
      float ov = __shfl_xor(bv, off, 32);
      int   oq = __shfl_xor(bq, off, 32);
      if (ov < bv || (ov == bv && oq < bq)) { bv = ov; bq = oq; }
    }
    #pragma unroll
    for (int i = 0; i < 10; ++i)
      if (bq == lane + 32 * i) v[i] = 3.0e38f;   // mark taken
    if (lane == 0) {
      outSrc[b * (Nt * MATCH) + nt * MATCH + j] = (float)bq;
      outTgt[b * (Nt * MATCH) + nt * MATCH + j] = (float)nt;
    }
  }
}

// ---------------- K5: dedup -> min_idx + src_mask -------------------
__global__ __launch_bounds__(320) void dedup_kernel(
    const float* __restrict__ cost, const float* __restrict__ outSrc,
    float* __restrict__ outMin, float* __restrict__ outMask) {
  __shared__ int ssrc[Nt * MATCH];
  int b = blockIdx.x;
  int tid = threadIdx.x;
  if (tid < Nt * MATCH) ssrc[tid] = (int)(outSrc[b * Nt * MATCH + tid] + 0.5f);
  __syncthreads();
  if (tid >= Nq) return;
  int q = tid;
  float best = INFC;
  int bidx = 0, found = 0;
  for (int k = 0; k < Nt * MATCH; ++k) {
    if (ssrc[k] == q) {
      int nt = k / MATCH;
      float cv = cost[(size_t)(b * Nq + q) * Nt + nt];
      found = 1;
      if (cv < best) { best = cv; bidx = nt; }  // strict < => lowest nt on ties
    }
  }
  outMin[b * Nq + q]  = (float)bidx;
  outMask[b * Nq + q] = found ? 1.0f : 0.0f;
}

extern "C" void kernel_launch(void* const* d_in, const int* in_sizes, int n_in,
                              void* d_out, int out_size, void* d_ws, size_t ws_size,
                              hipStream_t stream) {
  (void)in_sizes; (void)n_in; (void)out_size; (void)ws_size;
  const float* pred_logits = (const float*)d_in[0];  // (B,Nq,P,1)
  const float* pred_ctrl   = (const float*)d_in[1];  // (B,Nq,P,2)
  const float* pred_text   = (const float*)d_in[2];  // (B,Nq,T,V1)
  const float* tgt_ctrl    = (const float*)d_in[3];  // (B,Nt,P,2)
  const int*   tgt_texts   = (const int*)d_in[4];    // (B,Nt,T)
  float* out = (float*)d_out;

  float* wsf  = (float*)d_ws;
  float* cls  = wsf;              // B*Nq floats
  float* logT = wsf + 1280;       // B*T*V1*Nq floats (~11.6 MB), aligned slot

  hipLaunchKernelGGL(class_cost_kernel, dim3((B * Nq + 255) / 256), dim3(256),
                     0, stream, pred_logits, cls);
  hipLaunchKernelGGL(logsoftmax_wmma_kernel,
                     dim3((B * Nq * T + 127) / 128), dim3(256), 0, stream,
                     pred_text, logT);
  hipLaunchKernelGGL(ctc_cost_kernel, dim3(B * Nt), dim3(320), 0, stream,
                     logT, cls, pred_ctrl, tgt_ctrl, tgt_texts, out + OUT_COST);
  hipLaunchKernelGGL(topk_kernel, dim3(32), dim3(128), 0, stream,
                     out + OUT_COST, out + OUT_SRC, out + OUT_TGT);
  hipLaunchKernelGGL(dedup_kernel, dim3(B), dim3(320), 0, stream,
                     out + OUT_COST, out + OUT_SRC, out + OUT_MIN, out + OUT_MASK);
}